// DDKFLayer_10969346474729
// MI455X (gfx1250) — compile-verified
//
#include <hip/hip_runtime.h>
#include <hip/hip_bf16.h>
#include <math.h>

// ---------------------------------------------------------------------------
// DDKF layer for MI455X (gfx1250, wave32).
//
// Key transforms vs the reference:
//  * windowed FFT  -> (B*NW x 20) x (20 x N) complex GEMM on v_wmma_f32_16x16x4_f32
//  * complement FFT -> FFT(K) - FFT(window)   (linearity; one DFT per batch row)
//  * single fused kernel per (b, 16-window stripe): pass A computes row maxima
//    (LDS reduction), pass B recomputes M/M1 in registers and writes the final
//    result once -> only 132.7 MB of HBM writes instead of ~530 MB of
//    intermediate traffic.  Recompute on the matrix pipe beats HBM round-trips.
// ---------------------------------------------------------------------------

typedef __attribute__((ext_vector_type(2))) float v2f;
typedef __attribute__((ext_vector_type(8))) float v8f;

#define BB      32
#define LL      512
#define NN      2045      // interpolated length
#define NPAD    2048      // padded stride
#define NW      507       // number of windows
#define WIN     20
#define TWO_PI  6.28318530717958647692f

// ---------------------------------------------------------------------------
// Kernel 1: cubic (Keys a=-0.75) interpolation + gamma-weighted kernels -> Kmat
// t_p = 0.25*p exactly, so f in {0,.25,.5,.75} and i0 = p>>2.
// ---------------------------------------------------------------------------
__global__ __launch_bounds__(256) void ddkf_kmat(const float* __restrict__ sig,
                                                 const float* __restrict__ gl,
                                                 float* __restrict__ Kmat) {
  int gid = blockIdx.x * 256 + threadIdx.x;
  if (gid >= BB * NPAD) return;
  int b = gid >> 11;
  int p = gid & (NPAD - 1);
  float val = 0.0f;
  if (p < NN) {
    const float a = -0.75f;
    int i0 = p >> 2;
    float f = 0.25f * (float)(p & 3);
    float fp1 = 1.0f + f, fm1 = 1.0f - f, fm2 = 2.0f - f;
    float w0 = a*fp1*fp1*fp1 - 5.0f*a*fp1*fp1 + 8.0f*a*fp1 - 4.0f*a;
    float w1 = (a+2.0f)*f*f*f - (a+3.0f)*f*f + 1.0f;
    float w2 = (a+2.0f)*fm1*fm1*fm1 - (a+3.0f)*fm1*fm1 + 1.0f;
    float w3 = a*fm2*fm2*fm2 - 5.0f*a*fm2*fm2 + 8.0f*a*fm2 - 4.0f*a;
    int iA = (i0 - 1 < 0) ? 0 : i0 - 1;
    int iB = i0;
    int iC = (i0 + 1 > LL-1) ? LL-1 : i0 + 1;
    int iD = (i0 + 2 > LL-1) ? LL-1 : i0 + 2;
    const float* srow = sig + b * LL;
    float x = w0*srow[iA] + w1*srow[iB] + w2*srow[iC] + w3*srow[iD];
    float g0 = gl[0], g1 = gl[1];
    float e0 = expf(g0), e1 = expf(g1);
    float inv = 1.0f / (e0 + e1);
    float gam0 = e0 * inv, gam1 = e1 * inv;
    float xp = x + 1.3f;
    float d  = x - 0.7f;
    val = gam0 * (xp * xp) + gam1 * expf(-0.5f * d * d);
  }
  Kmat[gid] = val;
}

// ---------------------------------------------------------------------------
// Kernel 2a: twiddle tables.
//   Wc/Ws[t][p] = e^{-2*pi*i*t*p/N}  (t<20, DFT of the 20-tap window)
//   rotc/rots[m] = e^{-2*pi*i*m/N}   (window-start phase rotation, FK twiddles)
// ---------------------------------------------------------------------------
__global__ __launch_bounds__(256) void ddkf_tables(float* __restrict__ Wc,
                                                   float* __restrict__ Ws,
                                                   float* __restrict__ rotc,
                                                   float* __restrict__ rots) {
  int gid = blockIdx.x * 256 + threadIdx.x;   // 21*2048 total
  const float w = TWO_PI / (float)NN;
  if (gid < WIN * NPAD) {
    int t = gid >> 11, p = gid & (NPAD - 1);
    unsigned m = ((unsigned)(t * p)) % (unsigned)NN;
    float s, c;
    sincosf(w * (float)m, &s, &c);
    Wc[gid] = c;
    Ws[gid] = -s;
  } else if (gid < (WIN + 1) * NPAD) {
    int mm = gid - WIN * NPAD;
    unsigned m = (unsigned)mm % (unsigned)NN;
    float s, c;
    sincosf(w * (float)m, &s, &c);
    rotc[mm] = c;
    rots[mm] = -s;
  }
}

// ---------------------------------------------------------------------------
// Kernel 2b: FK[b,p] = full DFT of K row (one block per batch row, LDS-staged).
// Incremental (j*p mod N) avoids any fp phase error.
// ---------------------------------------------------------------------------
__global__ __launch_bounds__(256) void ddkf_fk(const float* __restrict__ Kmat,
                                               const float* __restrict__ rotc,
                                               const float* __restrict__ rots,
                                               float* __restrict__ FKre,
                                               float* __restrict__ FKim) {
  __shared__ float s_k[NPAD], s_rc[NPAD], s_rs[NPAD];
  int b = blockIdx.x, tid = threadIdx.x;
  for (int i = tid; i < NPAD; i += 256) {
    s_k[i]  = Kmat[b * NPAD + i];
    s_rc[i] = rotc[i];
    s_rs[i] = rots[i];
  }
  __syncthreads();
  for (int p = tid; p < NPAD; p += 256) {
    float ar = 0.0f, ai = 0.0f;
    if (p < NN) {
      int m = 0;
      for (int j = 0; j < NN; ++j) {
        float k = s_k[j];
        ar = fmaf(k, s_rc[m], ar);
        ai = fmaf(k, s_rs[m], ai);
        m += p;
        if (m >= NN) m -= NN;
      }
    }
    FKre[b * NPAD + p] = ar;
    FKim[b * NPAD + p] = ai;
  }
}

// ---------------------------------------------------------------------------
// WMMA tile: G[16 windows x 16 freqs] = A(16x20 window samples) x W(20x16).
// f32 16x16x4, K=20 as 5 steps.  Layouts per ISA 7.12.2:
//   A 16x4: lane l<16 holds A[l][t0+0..1] in v0..v1, lane 16+l holds A[l][t0+2..3]
//   B 4x16: lane l<16 holds B[t0+0..1][l],           lane 16+l holds B[t0+2..3][l]
//   C/D:    vgpr r -> row r (lanes 0-15) / row r+8 (lanes 16-31), col = lane%16
// ---------------------------------------------------------------------------
__device__ __forceinline__ void wmma_tile_dft(const float* s_kwin,
                                              const float* __restrict__ Wc,
                                              const float* __restrict__ Ws,
                                              int p0, int l, int h,
                                              v8f& cr, v8f& ci) {
  cr = (v8f){0.f,0.f,0.f,0.f,0.f,0.f,0.f,0.f};
  ci = (v8f){0.f,0.f,0.f,0.f,0.f,0.f,0.f,0.f};
#pragma unroll
  for (int t0 = 0; t0 < WIN; t0 += 4) {
    int ka = t0 + 2 * h;                 // this lane's K-pair within the step
    v2f a, bc, bs;
    a.x  = s_kwin[4 * l + ka];
    a.y  = s_kwin[4 * l + ka + 1];
    bc.x = Wc[ka * NPAD + p0 + l];
    bc.y = Wc[(ka + 1) * NPAD + p0 + l];
    bs.x = Ws[ka * NPAD + p0 + l];
    bs.y = Ws[(ka + 1) * NPAD + p0 + l];
    cr = __builtin_amdgcn_wmma_f32_16x16x4_f32(false, a, false, bc, (short)0, cr, false, false);
    ci = __builtin_amdgcn_wmma_f32_16x16x4_f32(false, a, false, bs, (short)0, ci, false, false);
  }
}

// ---------------------------------------------------------------------------
// Kernel 3: fused main kernel.  Block = (b, stripe of 16 windows), 8 waves.
// Pass A: WMMA DFT -> |Fwin| -> per-row max (LDS atomic-max, values >= 0).
// Pass B: recompute, M1 = |FK - Fwin|, threshold, min, single HBM write.
// ---------------------------------------------------------------------------
__global__ __launch_bounds__(256) void ddkf_main(const float* __restrict__ Kmat,
                                                 const float* __restrict__ Wc,
                                                 const float* __restrict__ Ws,
                                                 const float* __restrict__ rotc,
                                                 const float* __restrict__ rots,
                                                 const float* __restrict__ FKre,
                                                 const float* __restrict__ FKim,
                                                 const float* __restrict__ beta,
                                                 float* __restrict__ out) {
  __shared__ float s_rotc[NPAD], s_rots[NPAD];
  __shared__ float s_fkre[NPAD], s_fkim[NPAD];
  __shared__ float s_kwin[96];
  __shared__ float s_rowmax[16];
  __shared__ float s_thresh[16];

  const int tid = threadIdx.x;
  const int b   = blockIdx.y;
  const int w0  = blockIdx.x * 16;     // first window of this stripe
  const int j0  = 4 * w0;              // first sample touched by the stripe

  for (int i = tid; i < NPAD; i += 256) {
    s_rotc[i] = rotc[i];
    s_rots[i] = rots[i];
    s_fkre[i] = (i < NN) ? FKre[b * NPAD + i] : 0.0f;
    s_fkim[i] = (i < NN) ? FKim[b * NPAD + i] : 0.0f;
  }
  if (tid < 96) {
    int j = j0 + tid;
    s_kwin[tid] = (tid < 80 && j < NN) ? Kmat[b * NPAD + j] : 0.0f;
  }
  if (tid < 16) s_rowmax[tid] = 0.0f;
  __syncthreads();

  const int l    = tid & 15;
  const int h    = (tid >> 4) & 1;
  const int wave = tid >> 5;

  // ---- pass A: row maxima of M ------------------------------------------
  float pmax[8];
#pragma unroll
  for (int r = 0; r < 8; ++r) pmax[r] = 0.0f;

  for (int tile = wave; tile < NPAD / 16; tile += 8) {
    int p0 = tile * 16;
    v8f cr, ci;
    wmma_tile_dft(s_kwin, Wc, Ws, p0, l, h, cr, ci);
    int p = p0 + l;
    bool pv = (p < NN);
#pragma unroll
    for (int r = 0; r < 8; ++r) {
      int w = w0 + r + 8 * h;
      unsigned m = ((unsigned)(4 * w) * (unsigned)p) % (unsigned)NN;
      float Rc = s_rotc[m], Rs = s_rots[m];
      float Gr = cr[r], Gi = ci[r];
      float Fr = Rc * Gr - Rs * Gi;
      float Fi = Rc * Gi + Rs * Gr;
      float Mv = sqrtf(Fr * Fr + Fi * Fi);
      pmax[r] = fmaxf(pmax[r], pv ? Mv : 0.0f);
    }
  }
#pragma unroll
  for (int r = 0; r < 8; ++r)
    atomicMax((int*)&s_rowmax[r + 8 * h], __float_as_int(pmax[r]));  // vals >= 0
  __syncthreads();
  if (tid < 16) s_thresh[tid] = s_rowmax[tid] * beta[0];
  __syncthreads();

  // ---- pass B: recompute, threshold, write -------------------------------
  for (int tile = wave; tile < NPAD / 16; tile += 8) {
    int p0 = tile * 16;
    v8f cr, ci;
    wmma_tile_dft(s_kwin, Wc, Ws, p0, l, h, cr, ci);
    int p = p0 + l;
#pragma unroll
    for (int r = 0; r < 8; ++r) {
      int row = r + 8 * h;
      int w = w0 + row;
      unsigned m = ((unsigned)(4 * w) * (unsigned)p) % (unsigned)NN;
      float Rc = s_rotc[m], Rs = s_rots[m];
      float Gr = cr[r], Gi = ci[r];
      float Fr = Rc * Gr - Rs * Gi;
      float Fi = Rc * Gi + Rs * Gr;
      float Mv = sqrtf(Fr * Fr + Fi * Fi);
      float Cr = s_fkre[p] - Fr;
      float Cm = s_fkim[p] - Fi;
      float M1 = sqrtf(Cr * Cr + Cm * Cm);
      float strong = (Mv > s_thresh[row]) ? 1.0f : 0.0f;
      float outv = fminf(Mv, M1 * Mv * strong);
      if (w < NW && p < NN)
        out[((size_t)b * NW + w) * NN + p] = outv;
    }
  }
}

// ---------------------------------------------------------------------------
// Launcher.  Workspace layout (floats):
//   Kmat 32x2048 | Wc 20x2048 | Ws 20x2048 | rotc 2048 | rots 2048 |
//   FKre 32x2048 | FKim 32x2048   -> 282,624 floats (~1.1 MB)
// ---------------------------------------------------------------------------
extern "C" void kernel_launch(void* const* d_in, const int* in_sizes, int n_in,
                              void* d_out, int out_size, void* d_ws, size_t ws_size,
                              hipStream_t stream) {
  const float* signal = (const float*)d_in[0];
  // d_in[1] = alpha (unused by the reference forward pass)
  const float* beta   = (const float*)d_in[2];
  const float* glog   = (const float*)d_in[3];
  float* out = (float*)d_out;

  float* ws   = (float*)d_ws;
  float* Kmat = ws;                       // 65536
  float* Wc   = Kmat + BB * NPAD;         // 40960
  float* Ws_  = Wc + WIN * NPAD;          // 40960
  float* rotc = Ws_ + WIN * NPAD;         // 2048
  float* rots = rotc + NPAD;              // 2048
  float* FKre = rots + NPAD;              // 65536
  float* FKim = FKre + BB * NPAD;         // 65536

  ddkf_kmat<<<(BB * NPAD) / 256, 256, 0, stream>>>(signal, glog, Kmat);
  ddkf_tables<<<((WIN + 1) * NPAD) / 256, 256, 0, stream>>>(Wc, Ws_, rotc, rots);
  ddkf_fk<<<BB, 256, 0, stream>>>(Kmat, rotc, rots, FKre, FKim);

  dim3 grid(32 /* ceil(507/16) */, BB);
  ddkf_main<<<grid, 256, 0, stream>>>(Kmat, Wc, Ws_, rotc, rots, FKre, FKim, beta, out);
}